// ScharrAndLaplaceLoss_54571854463846
// MI455X (gfx1250) — compile-verified
//
#include <hip/hip_runtime.h>
#include <hip/hip_bf16.h>
#include <stdint.h>

// Problem constants (fixed by the reference: B=32, H=480, W=640, fp32).
#define BATCH 32
#define IH    480
#define IW    640
#define TW    64          // output tile width  (640 = 10 * 64)
#define THT   32          // output tile height (480 = 15 * 32)
#define RW    (TW + 2)    // staged region width  (halo of 1)
#define RH    (THT + 2)   // staged region height (halo of 1)
#define LDP   68          // LDS row pitch in floats (mult of 4, != mult of 64)
#define NPIX  ((float)BATCH * (float)IH * (float)IW)

typedef float v2f_t __attribute__((ext_vector_type(2)));
typedef float v8f_t __attribute__((ext_vector_type(8)));

// ---- CDNA5 async global->LDS staging (ASYNCcnt path), guarded ------------
#if defined(__AMDGCN__) && __has_builtin(__builtin_amdgcn_global_load_async_to_lds_b32)
#define USE_ASYNC_LDS 1
typedef __attribute__((address_space(1))) int as1_int;   // clang prints as '__device__ int *'
typedef __attribute__((address_space(3))) int as3_int;   // clang prints as '__shared__ int *'
__device__ __forceinline__ void async_g2l_b32(const float* gp, float* lp) {
  // Generic->AS casts via integer round trip: AS3 address is the low 32 bits
  // of the flat LDS-aperture address.
  __builtin_amdgcn_global_load_async_to_lds_b32(
      (as1_int*)(unsigned long long)gp,
      (as3_int*)(unsigned int)(unsigned long long)lp,
      0, 0);
}
__device__ __forceinline__ void wait_asynccnt0() {
#if __has_builtin(__builtin_amdgcn_s_wait_asynccnt)
  __builtin_amdgcn_s_wait_asynccnt(0);
#else
  asm volatile("s_wait_asynccnt 0" ::: "memory");
#endif
}
#endif

// ---- WMMA-based block reduction, guarded ---------------------------------
#if defined(__AMDGCN__) && __has_builtin(__builtin_amdgcn_wmma_f32_16x16x4_f32)
#define USE_WMMA_RED 1
#endif

__global__ void ScharrLaplace_zero_ws(float* ws) { ws[0] = 0.0f; }

__global__ void ScharrLaplace_finalize(const float* __restrict__ ws,
                                       float* __restrict__ out) {
  out[0] = ws[0] * (1.0f / NPIX);
}

__global__ __launch_bounds__(256)
void ScharrLaplace_tile_kernel(const float* __restrict__ pred,
                               const float* __restrict__ target,
                               float* __restrict__ accum) {
  __shared__ __align__(16) float sdp[RH * LDP];  // pred tile, then diff
  __shared__ __align__(16) float sdt[RH * LDP];  // target tile
  __shared__ float red[256];

  const int tid = threadIdx.x;
  const int tx = blockIdx.x, ty = blockIdx.y, b = blockIdx.z;
  const int gr0 = ty * THT - 1;   // top-left of halo'd region in image coords
  const int gc0 = tx * TW - 1;
  const float* __restrict__ pplane = pred   + (size_t)b * (IH * IW);
  const float* __restrict__ tplane = target + (size_t)b * (IH * IW);

  // --- Stage halo'd tiles of pred & target into LDS (zero-pad OOB) -------
  for (int idx = tid; idx < RH * RW; idx += 256) {
    const int r  = idx / RW;
    const int c  = idx - r * RW;
    const int gr = gr0 + r;
    const int gc = gc0 + c;
    const int l  = r * LDP + c;
    if ((unsigned)gr < (unsigned)IH && (unsigned)gc < (unsigned)IW) {
      const int g = gr * IW + gc;
#if USE_ASYNC_LDS
      async_g2l_b32(pplane + g, &sdp[l]);
      async_g2l_b32(tplane + g, &sdt[l]);
#else
      sdp[l] = pplane[g];
      sdt[l] = tplane[g];
#endif
    } else {
      sdp[l] = 0.0f;
      sdt[l] = 0.0f;
    }
  }
#if USE_ASYNC_LDS
  wait_asynccnt0();
#endif
  __syncthreads();

  // --- Linearity: d = pred - target, in place (padding cols never read) --
  for (int i = tid; i < RH * LDP; i += 256) sdp[i] -= sdt[i];
  __syncthreads();

  // --- Register-blocked stencil: 2 rows x 4 cols per thread ---------------
  const int cg = tid & 15;     // 16 column groups * 4 cols = 64
  const int rg = tid >> 4;     // 16 row groups    * 2 rows = 32
  const int r0 = rg * 2;
  const int c0 = cg * 4;

  float n[4][6];
#pragma unroll
  for (int rr = 0; rr < 4; ++rr) {
    const float* row = &sdp[(r0 + rr) * LDP + c0];  // 16B aligned
    const float4 q  = *(const float4*)row;
    const float2 q2 = *(const float2*)(row + 4);
    n[rr][0] = q.x;  n[rr][1] = q.y;  n[rr][2] = q.z;  n[rr][3] = q.w;
    n[rr][4] = q2.x; n[rr][5] = q2.y;
  }

  float acc = 0.0f;
#pragma unroll
  for (int o = 0; o < 2; ++o) {
#pragma unroll
    for (int j = 0; j < 4; ++j) {
      const float a00 = n[o][j],     a01 = n[o][j + 1], a02 = n[o][j + 2];
      const float a10 = n[o + 1][j], a11 = n[o + 1][j + 1], a12 = n[o + 1][j + 2];
      const float a20 = n[o + 2][j], a21 = n[o + 2][j + 1], a22 = n[o + 2][j + 2];
      // cross-correlation with the reference kernels (incl. the -3 quirk)
      const float sx = -3.f * a00 - 3.f * a02 - 10.f * a10 + 10.f * a12
                       - 3.f * a20 + 3.f * a22;
      const float sy = -3.f * a00 - 10.f * a01 - 3.f * a02
                       + 3.f * a20 + 10.f * a21 + 3.f * a22;
      const float lp = a01 + a10 - 4.f * a11 + a12 + a21;
      acc += fabsf(sx) + fabsf(sy) + fabsf(lp);
    }
  }

  // --- Block reduction: LDS -> wave0 WMMA(ones x B) -> atomic -------------
  red[tid] = acc;
  __syncthreads();

  if (tid < 32) {   // exactly wave 0 (wave32): EXEC all-ones, WMMA-legal
#if USE_WMMA_RED
    v2f_t a; a[0] = 1.0f; a[1] = 1.0f;   // A = ones(16x4)
    v8f_t c = {};
    v2f_t bv;
    bv[0] = red[tid];       bv[1] = red[tid + 32];
    c = __builtin_amdgcn_wmma_f32_16x16x4_f32(false, a, false, bv, (short)0, c, false, false);
    bv[0] = red[tid + 64];  bv[1] = red[tid + 96];
    c = __builtin_amdgcn_wmma_f32_16x16x4_f32(false, a, false, bv, (short)0, c, false, false);
    bv[0] = red[tid + 128]; bv[1] = red[tid + 160];
    c = __builtin_amdgcn_wmma_f32_16x16x4_f32(false, a, false, bv, (short)0, c, false, false);
    bv[0] = red[tid + 192]; bv[1] = red[tid + 224];
    c = __builtin_amdgcn_wmma_f32_16x16x4_f32(false, a, false, bv, (short)0, c, false, false);
    // D[m][n] = column sums; lanes 0..15 carry row M=0, lanes 16..31 row M=8
    // (identical rows), so summing c[0] over all 32 lanes gives 2 * total.
    float s = c[0];
#pragma unroll
    for (int off = 16; off > 0; off >>= 1) s += __shfl_xor(s, off, 32);
    s *= 0.5f;
#else
    float s = 0.0f;
#pragma unroll
    for (int i = 0; i < 8; ++i) s += red[tid + 32 * i];
#pragma unroll
    for (int off = 16; off > 0; off >>= 1) s += __shfl_xor(s, off, 32);
#endif
    if (tid == 0) atomicAdd(accum, s);
  }
}

extern "C" void kernel_launch(void* const* d_in, const int* in_sizes, int n_in,
                              void* d_out, int out_size, void* d_ws, size_t ws_size,
                              hipStream_t stream) {
  (void)in_sizes; (void)n_in; (void)out_size; (void)ws_size;
  const float* pred   = (const float*)d_in[0];
  const float* target = (const float*)d_in[1];
  float* out = (float*)d_out;
  float* ws  = (float*)d_ws;

  ScharrLaplace_zero_ws<<<dim3(1), dim3(1), 0, stream>>>(ws);

  dim3 grid(IW / TW, IH / THT, BATCH);   // 10 x 15 x 32 = 4800 blocks
  ScharrLaplace_tile_kernel<<<grid, dim3(256), 0, stream>>>(pred, target, ws);

  ScharrLaplace_finalize<<<dim3(1), dim3(1), 0, stream>>>(ws, out);
}